// TTLinear_84267258347697
// MI455X (gfx1250) — compile-verified
//
#include <hip/hip_runtime.h>

typedef float v2f __attribute__((ext_vector_type(2)));
typedef float v8f __attribute__((ext_vector_type(8)));

#define K_DIM   8192      // IN_SHAPE[0]*IN_SHAPE[1]
#define N_DIM   16        // OUT_SHAPE[0]*OUT_SHAPE[1]
#define BATCH_  16384
#define NGROUPS (K_DIM / 4)   // K-groups of 4 per V_WMMA_F32_16X16X4_F32

// ---------------------------------------------------------------------------
// Kernel 1: reconstruct W from TT cores, storing it pre-swizzled into the
// WMMA B-fragment layout (4x16 f32 B, 2 VGPRs/lane):
//   v0: lanes 0-15 = row K=0 (N=lane), lanes 16-31 = row K=2
//   v1: lanes 0-15 = row K=1,          lanes 16-31 = row K=3
// For K-group g, lane l reads float2 at wfrag[g*64 + l*2].
// ---------------------------------------------------------------------------
__global__ void tt_build_wfrag(const float* __restrict__ core0,   // (1,128,4,8)
                               const float* __restrict__ core1,   // (8,64,4,1)
                               float* __restrict__ wfrag) {
  int idx = blockIdx.x * blockDim.x + threadIdx.x;   // 0 .. 8192*16-1
  if (idx >= K_DIM * N_DIM) return;
  int k = idx >> 4;          // 0..8191  (= i*64 + j)
  int n = idx & 15;          // 0..15    (= o*4 + p)
  int i = k >> 6, j = k & 63;
  int o = n >> 2, p = n & 3;
  float acc = 0.f;
#pragma unroll
  for (int r = 0; r < 8; ++r)
    acc += core0[i * 32 + o * 8 + r] * core1[r * 256 + j * 4 + p];
  int g    = k >> 2;
  int kr   = k & 3;
  int lane = ((kr >> 1) << 4) + n;   // K=2,3 live in lanes 16-31
  int slot = kr & 1;                 // K=1,3 live in v1
  wfrag[g * 64 + lane * 2 + slot] = acc;
}

// ---------------------------------------------------------------------------
// Kernel 2: out = x @ W via V_WMMA_F32_16X16X4_F32.
// One wave per 16-row tile of x; K swept in groups of 4; 8 groups unrolled per
// iteration with 4 round-robin accumulators to break the D->C chain.
// ---------------------------------------------------------------------------
__global__ __launch_bounds__(256) void tt_gemm_wmma(
    const float* __restrict__ x,      // (16384, 8192) row-major
    const float* __restrict__ wfrag,  // B fragments, L2-resident (512 KB)
    float* __restrict__ out) {        // (16384, 16) row-major
  const int lane = threadIdx.x & 31;
  const int wave = threadIdx.x >> 5;
  const int tile = blockIdx.x * 8 + wave;          // 0..1023

  // A-fragment base: 16x4 f32 A layout
  //   v0 = x[row][kbase + 2*(lane>>4)], v1 = next column.
  const int row = (tile << 4) + (lane & 15);
  const float* ap = x + (size_t)row * K_DIM + ((lane >> 4) << 1);
  const float* bp = wfrag + lane * 2;

  v8f c0 = {0.f, 0.f, 0.f, 0.f, 0.f, 0.f, 0.f, 0.f};
  v8f c1 = c0, c2 = c0, c3 = c0;

  for (int g = 0; g < NGROUPS; g += 8) {
    v2f a[8], b[8];
#pragma unroll
    for (int u = 0; u < 8; ++u) {
      // x is streamed exactly once: non-temporal so W stays hot in L2.
      a[u] = __builtin_nontemporal_load((const v2f*)(ap + (size_t)(g + u) * 4));
      b[u] = *(const v2f*)(bp + (size_t)(g + u) * 64);
    }
#pragma unroll
    for (int u = 0; u < 8; ++u) {
      v8f& acc = (u & 3) == 0 ? c0 : (u & 3) == 1 ? c1 : (u & 3) == 2 ? c2 : c3;
      // (neg_a, A, neg_b, B, c_mod, C, reuse_a, reuse_b)
      acc = __builtin_amdgcn_wmma_f32_16x16x4_f32(
          false, a[u], false, b[u], (short)0, acc, false, false);
    }
  }

  v8f c = (c0 + c1) + (c2 + c3);

  // D layout (16x16 f32 C/D): VGPR r -> M = r + 8*(lane>>4), N = lane&15.
  const int m0 = (tile << 4) + ((lane >> 4) << 3);
  float* op = out + (size_t)m0 * N_DIM + (lane & 15);
#pragma unroll
  for (int r = 0; r < 8; ++r)
    op[r * N_DIM] = c[r];
}

// ---------------------------------------------------------------------------
// Host-side launcher (graph-capture safe: only kernel launches on `stream`).
// Inputs: d_in[0]=x (f32, 16384*8192), d_in[1]=core0 (f32, 4096),
//         d_in[2]=core1 (f32, 2048). Output: f32, 16384*16.
// Workspace: 512 KB for the pre-swizzled W fragments.
// ---------------------------------------------------------------------------
extern "C" void kernel_launch(void* const* d_in, const int* in_sizes, int n_in,
                              void* d_out, int out_size, void* d_ws, size_t ws_size,
                              hipStream_t stream) {
  const float* x     = (const float*)d_in[0];
  const float* core0 = (const float*)d_in[1];
  const float* core1 = (const float*)d_in[2];
  float* out   = (float*)d_out;
  float* wfrag = (float*)d_ws;   // needs NGROUPS*64*4 = 512 KB

  // Stage 1: build W in B-fragment order (131072 elements).
  tt_build_wfrag<<<(K_DIM * N_DIM + 255) / 256, 256, 0, stream>>>(core0, core1, wfrag);

  // Stage 2: WMMA GEMM. 1024 tiles of 16 rows; 8 waves per 256-thread block.
  tt_gemm_wmma<<<BATCH_ / 16 / 8, 256, 0, stream>>>(x, wfrag, out);
}